// Circuit_88871463288913
// MI455X (gfx1250) — compile-verified
//
#include <hip/hip_runtime.h>
#include <math.h>

typedef float v2f __attribute__((ext_vector_type(2)));
typedef float v8f __attribute__((ext_vector_type(8)));

#define D_TOT 2048
#define B_TOT 32
#define BC    2                    /* batch columns per workgroup            */
#define NWG   (B_TOT / BC)         /* 16 workgroups                          */
#define NTHR  512                  /* 16 wave32 per workgroup                */
#define COLS  ((D_TOT / 16) * BC)  /* 256 matmul columns per WG per pass     */

// ---------------------------------------------------------------------------
// Pass schedule: 11 chunks of CONSECUTIVE reference gates, each inside a
// 4-qubit window [w, w+3].  Window position p maps to group-bit (3-p):
// qubit w is the most-significant bit of the 4-bit group index g.
// ---------------------------------------------------------------------------
__device__ const int c_pass_w[11] = {0, 4, 7, 0, 3, 6, 7, 0, 3, 6, 7};
__device__ const int c_ng[11]     = {4, 4, 3, 3, 3, 3, 1, 3, 3, 3, 1};
__device__ const int c_gstart[11] = {0, 4, 8, 11, 14, 17, 20, 21, 24, 27, 30};
// gate type: 0 = H (single qubit), 1 = CNOT(p, p+1), 2 = CRX(p, p+1)
__device__ const int c_gtype[31] = {0,0,0,0, 0,0,0,0, 0,0,0,
                                    1,1,1, 1,1,1, 1,1,1, 1,
                                    2,2,2, 2,2,2, 2,2,2, 2};
__device__ const int c_gpos[31]  = {0,1,2,3, 0,1,2,3, 1,2,3,
                                    0,1,2, 0,1,2, 0,1,2, 2,
                                    0,1,2, 0,1,2, 0,1,2, 2};
__device__ const int c_gang[31]  = {0,0,0,0, 0,0,0,0, 0,0,0,
                                    0,0,0, 0,0,0, 0,0,0, 0,
                                    0,1,2, 3,4,5, 6,7,8, 9};

// ---------------------------------------------------------------------------
// Fully fused circuit kernel.
//   Phase 1: build 11 pass unitaries (16x16 complex) in LDS.
//   Phase 2: load this WG's batch slice (2048 x BC complex) into LDS.
//   Phase 3: 11 in-place WMMA passes over LDS, barrier between passes.
//   Phase 4: store interleaved complex64 to d_out.
// WMMA fragment layouts per CDNA5 ISA 7.12.2:
//   A 16x4 : lane M = lane&15 ; VGPR0 K = 2*half, VGPR1 K = 2*half+1
//   B 4x16 : lane N = lane&15 ; VGPR0 K = 2*half, VGPR1 K = 2*half+1
//   C/D    : VGPR v holds row (v + 8*half), col lane&15
// ---------------------------------------------------------------------------
__global__ void __launch_bounds__(NTHR)
circuit_fused(const float* __restrict__ xre,
              const float* __restrict__ xim,
              const float* __restrict__ angles,
              float* __restrict__ out) {
  __shared__ float lsre[D_TOT * BC];   // 16 KB
  __shared__ float lsim[D_TOT * BC];   // 16 KB
  __shared__ float mats[11 * 512];     // 22 KB: per pass [re 256][im 256]

  const int tid  = threadIdx.x;
  const int b0   = blockIdx.x * BC;

  // ---------------- Phase 1: build pass unitaries ----------------
  {
    const int e = tid;               // element id (first 256 threads active)
    const int r = e >> 4;
    const int c = e & 15;
    for (int p = 0; p < 11; ++p) {
      if (e < 256) {
        mats[p * 512 + e]       = (r == c) ? 1.0f : 0.0f;
        mats[p * 512 + 256 + e] = 0.0f;
      }
      __syncthreads();
      for (int gi = 0; gi < c_ng[p]; ++gi) {
        float sr = 0.0f, si = 0.0f;
        if (e < 256) {
          const int g   = c_gstart[p] + gi;
          const int ty  = c_gtype[g];
          const int pos = c_gpos[g];
          const float th = (ty == 2) ? angles[c_gang[g]] : 0.0f;
          // T[r][c] = sum_k G[r][k] * U[k][c]   (complex)
          for (int k = 0; k < 16; ++k) {
            float gre = 0.0f, gim = 0.0f;
            if (ty == 0) {                     // Hadamard at window pos
              const int bb = 3 - pos, m = 1 << bb, oth = 15 ^ m;
              if ((r & oth) == (k & oth)) {
                const float v = 0.70710678118654752f;
                gre = (((r >> bb) & 1) & ((k >> bb) & 1)) ? -v : v;
              }
            } else if (ty == 1) {              // CNOT
              const int cb = 3 - pos, tb = 2 - pos;
              const int exp_r = k ^ (((k >> cb) & 1) << tb);
              gre = (r == exp_r) ? 1.0f : 0.0f;
            } else {                           // CRX
              const int cb = 3 - pos, tb = 2 - pos;
              const int cd = (k >> cb) & 1;
              const float a = 0.5f * th * (float)cd;
              if (r == k)                    gre =  cosf(a);
              else if (r == (k ^ (1 << tb))) gim = -sinf(a);
            }
            const float ur = mats[p * 512 + ((k << 4) | c)];
            const float ui = mats[p * 512 + 256 + ((k << 4) | c)];
            sr += gre * ur - gim * ui;
            si += gre * ui + gim * ur;
          }
        }
        __syncthreads();                       // all reads of U done
        if (e < 256) {
          mats[p * 512 + e]       = sr;        // in-place U <- G @ U
          mats[p * 512 + 256 + e] = si;
        }
        __syncthreads();
      }
    }
  }

  // ---------------- Phase 2: load batch slice into LDS ----------------
#pragma unroll
  for (int k = 0; k < (D_TOT * BC) / NTHR; ++k) {
    const int f  = tid + k * NTHR;             // f = i*BC + bl
    const int i  = f >> 1;
    const int bl = f & 1;
    lsre[f] = xre[i * B_TOT + b0 + bl];
    lsim[f] = xim[i * B_TOT + b0 + bl];
  }
  __syncthreads();

  // ---------------- Phase 3: 11 in-place WMMA passes ----------------
  const int lane = tid & 31;
  const int wave = tid >> 5;                   // tile id, 0..15
  const int n    = lane & 15;
  const int half = lane >> 4;
  const int M    = n;                          // A-fragment row

  for (int p = 0; p < 11; ++p) {
    const int w      = c_pass_w[p];
    const int L      = 1 << (7 - w);           // stride of lowest window qubit
    const int shiftc = 8 - w;                  // log2(L * BC)
    const int c      = wave * 16 + n;          // this lane's column, 0..255
    const int hi     = c >> shiftc;
    const int rem    = c & ((1 << shiftc) - 1);
    const int lo     = rem >> 1;
    const int bl     = rem & 1;
    const int ibase  = hi * (L << 4) + lo;     // row index at g = 0

    v8f cr = {0.f, 0.f, 0.f, 0.f, 0.f, 0.f, 0.f, 0.f};
    v8f ci = {0.f, 0.f, 0.f, 0.f, 0.f, 0.f, 0.f, 0.f};

#pragma unroll
    for (int kc = 0; kc < 4; ++kc) {
      const int k0 = kc * 4 + 2 * half;
      const int k1 = k0 + 1;

      v2f ar, ai, ani, br, bi;
      ar[0]  = mats[p * 512 + M * 16 + k0];
      ar[1]  = mats[p * 512 + M * 16 + k1];
      ai[0]  = mats[p * 512 + 256 + M * 16 + k0];
      ai[1]  = mats[p * 512 + 256 + M * 16 + k1];
      ani[0] = -ai[0];
      ani[1] = -ai[1];

      const int a0 = (ibase + k0 * L) * BC + bl;
      const int a1 = (ibase + k1 * L) * BC + bl;
      br[0] = lsre[a0];
      br[1] = lsre[a1];
      bi[0] = lsim[a0];
      bi[1] = lsim[a1];

      // complex MAC: Cr += Ar*Br - Ai*Bi ; Ci += Ar*Bi + Ai*Br
      cr = __builtin_amdgcn_wmma_f32_16x16x4_f32(false, ar,  false, br, (short)0, cr, false, false);
      cr = __builtin_amdgcn_wmma_f32_16x16x4_f32(false, ani, false, bi, (short)0, cr, false, false);
      ci = __builtin_amdgcn_wmma_f32_16x16x4_f32(false, ar,  false, bi, (short)0, ci, false, false);
      ci = __builtin_amdgcn_wmma_f32_16x16x4_f32(false, ai,  false, br, (short)0, ci, false, false);
    }

    // In-place writeback: this wave exclusively owns (rows of its columns).
    // DS ops from one wave stay in order, so reads above precede these writes.
#pragma unroll
    for (int v = 0; v < 8; ++v) {
      const int m   = v + 8 * half;            // output row g'
      const int idx = (ibase + m * L) * BC + bl;
      lsre[idx] = cr[v];
      lsim[idx] = ci[v];
    }
    __syncthreads();                           // regroup before next pass
  }

  // ---------------- Phase 4: store interleaved complex64 ----------------
#pragma unroll
  for (int k = 0; k < (D_TOT * BC) / NTHR; ++k) {
    const int f  = tid + k * NTHR;
    const int i  = f >> 1;
    const int bl = f & 1;
    const int gi = i * B_TOT + b0 + bl;        // element index in (D, B)
    out[gi * 2]     = lsre[f];
    out[gi * 2 + 1] = lsim[f];
  }
}

// ---------------------------------------------------------------------------
extern "C" void kernel_launch(void* const* d_in, const int* in_sizes, int n_in,
                              void* d_out, int out_size, void* d_ws, size_t ws_size,
                              hipStream_t stream) {
  (void)in_sizes; (void)n_in; (void)out_size; (void)d_ws; (void)ws_size;

  const float* xre = (const float*)d_in[0];   // (2048, 32) f32
  const float* xim = (const float*)d_in[1];   // (2048, 32) f32
  const float* ang = (const float*)d_in[2];   // (10,) f32
  // d_in[3] = dim (2), d_in[4] = wires (11): hardcoded.

  circuit_fused<<<NWG, NTHR, 0, stream>>>(xre, xim, ang, (float*)d_out);
}